// DepthwiseXCorr_37538014167462
// MI455X (gfx1250) — compile-verified
//
#include <hip/hip_runtime.h>

// Depthwise x-corr on MI455X (gfx1250, wave32):
//   matrix math : V_WMMA_F32_16X16X4_F32 (fp32 in / fp32 acc, matches reference)
//   staging     : Tensor Data Mover (tensor_load_to_lds) global -> LDS, TENSORcnt
//
// Per channel ch (32768): out[y][x] = sum_{i<8,j<8} k[i][j]*s[y+i][x+j], out 25x25.
// GEMM form per 16x16 output tile (yb,xb in {0,16}):
//   A[m][j] = k[i][j-m]  (Toeplitz, zero outside 0<=j-m<8)  -> A operand 16x4
//   B[j][n] = s[yb+n+i][xb+j]                                -> B operand 4x16
//   D[m][n] = out[yb+n][xb+m], accumulated over i=0..7 and j-chunks of 4 (j=0..23)

typedef float v2f  __attribute__((ext_vector_type(2)));
typedef float v8f  __attribute__((ext_vector_type(8)));
typedef unsigned int u32x4 __attribute__((ext_vector_type(4)));
typedef int   i32x4 __attribute__((ext_vector_type(4)));
typedef int   i32x8 __attribute__((ext_vector_type(8)));

#define WAVES 4   // 4 waves per block, one channel per wave

__global__ __launch_bounds__(WAVES * 32)
void xcorr_dw_wmma(const float* __restrict__ kern,   // (nchan, 8, 8)
                   const float* __restrict__ srch,   // (nchan, 32, 32)
                   float* __restrict__ out,          // (nchan, 25, 25)
                   int nchan) {
  __shared__ __align__(16) float sS[WAVES][32 * 32];  // search tile per wave
  __shared__ float sT[WAVES][8 * 40];                 // zero-padded Toeplitz rows

  const int tid  = threadIdx.x;
  const int lane = tid & 31;
  const int wave = tid >> 5;
  const int ch   = blockIdx.x * WAVES + wave;

  float* S = sS[wave];
  float* T = sT[wave];

  if (ch < nchan) {
#if defined(__has_builtin) && __has_builtin(__builtin_amdgcn_tensor_load_to_lds)
    // ---- TDM: DMA the 32x32 fp32 search tile straight into LDS ----------
    // D# group 0: count=1 | lds_addr | global_addr[56:0] | type=2 ("image")
    const unsigned long long ga = (unsigned long long)(srch + (size_t)ch * 1024);
    const unsigned int lds_off  = (unsigned int)(size_t)S;   // LDS byte address
    u32x4 g0;
    g0.x = 0x1u;                                             // count=1, user D#
    g0.y = lds_off;                                          // lds_addr
    g0.z = (unsigned int)(ga & 0xFFFFFFFFull);               // global_addr lo
    g0.w = (unsigned int)((ga >> 32) & 0x01FFFFFFull) | (2u << 30); // hi | type=2
    // D# group 1: data_size=4B (code 2); tensor 32x32, tile 32x32, stride0=32,
    // stride1=1024 (unused for 2D); no multicast / barrier / pad / iterate.
    i32x8 g1;
    g1[0] = 2 << 16;                 // workgroup_mask=0, data_size=2
    g1[1] = 32 << 16;                // tensor_dim0 = 32 (bits 63:48)
    g1[2] = 32 << 16;                // tensor_dim0 hi=0 | tensor_dim1 = 32
    g1[3] = 32 << 16;                // tensor_dim1 hi=0 | tile_dim0 = 32
    g1[4] = 32;                      // tile_dim1 = 32, tile_dim2 = 0 (2D)
    g1[5] = 32;                      // tensor_dim0_stride = 32 (lo32)
    g1[6] = (int)(1024u << 16);      // stride0 hi=0 | tensor_dim1_stride lo16
    g1[7] = 0;                       // tensor_dim1_stride hi
    const i32x4 gz4 = {0, 0, 0, 0};  // groups 2/3 unused for a 2D tile
    const i32x8 gz8 = {0, 0, 0, 0, 0, 0, 0, 0};
    // amdgpu-toolchain (clang-23) arity: 6 args (extra i32x8 group, then cpol).
    __builtin_amdgcn_tensor_load_to_lds(g0, g1, gz4, gz4, gz8, 0);
#else
    // ---- fallback staging: b128 global -> VGPR -> LDS -------------------
    const float4* g4 = reinterpret_cast<const float4*>(srch + (size_t)ch * 1024);
    float4* s4 = reinterpret_cast<float4*>(S);
#pragma unroll
    for (int t = 0; t < 8; ++t) s4[t * 32 + lane] = g4[t * 32 + lane];
#endif

    // Toeplitz rows T[i][u], u in [0,40): k[i][u-15] for 15 <= u < 23, else 0.
    const float* kc = kern + (size_t)ch * 64;
    for (int t = lane; t < 8 * 40; t += 32) {
      const int i = t / 40;
      const int u = t - i * 40;
      float v = 0.0f;
      if (u >= 15 && u < 23) v = kc[i * 8 + (u - 15)];
      T[t] = v;
    }
  }

#if defined(__has_builtin) && __has_builtin(__builtin_amdgcn_s_wait_tensorcnt)
  __builtin_amdgcn_s_wait_tensorcnt(0);   // own tile resident in LDS
#endif
  __syncthreads();
  if (ch >= nchan) return;

  // f32 WMMA operand lane mapping (wave32):
  //   A 16x4 : lane L, reg r -> A[L&15][((L>>4)*2)+r]
  //   B 4x16 : lane L, reg r -> B[((L>>4)*2)+r][L&15]
  //   D 16x16: lane L, reg g -> D[g + (L>=16 ? 8 : 0)][L&15]
  const int m  = lane & 15;          // A row (x offset) == B col (y offset)
  const int kk = (lane >> 4) << 1;   // K-pair base for this half-wave

  v8f d00 = {}, d01 = {}, d10 = {}, d11 = {};   // tiles (yb,xb) in {0,16}^2

#pragma unroll 2
  for (int i = 0; i < 8; ++i) {
    const int r0 = m + i;                 // search row for yb=0 tiles (<= 22)
    int r1 = r0 + 16;                     // search row for yb=16 tiles
    if (r1 > 31) r1 = 31;                 // clamped rows feed garbage y>=25 only
#pragma unroll
    for (int c = 0; c < 6; ++c) {
      const int j0 = c * 4 + kk;          // global j of this lane's K pair

      // A operand: shifted kernel row (Toeplitz). Index 15 + j0 - m in [0,38].
      const int ta = i * 40 + 15 + j0 - m;
      v2f a; a.x = T[ta]; a.y = T[ta + 1];

      // B operand columns. xb=16 pair clamps with consecutive addresses so the
      // two reads fuse into one ds_load_b64; clamped values only ever multiply
      // structurally-zero Toeplitz entries for valid outputs (m <= 8).
      const int c00 = j0;                 // xb = 0, j <= 23 always in range
      int c10 = j0 + 16;                  // xb = 16
      if (c10 > 30) c10 = 30;

      const float* Sr0 = S + r0 * 32;
      const float* Sr1 = S + r1 * 32;
      v2f b;

      b.x = Sr0[c00]; b.y = Sr0[c00 + 1];
      d00 = __builtin_amdgcn_wmma_f32_16x16x4_f32(false, a, false, b, (short)0, d00, false, false);
      b.x = Sr0[c10]; b.y = Sr0[c10 + 1];
      d01 = __builtin_amdgcn_wmma_f32_16x16x4_f32(false, a, false, b, (short)0, d01, false, false);
      b.x = Sr1[c00]; b.y = Sr1[c00 + 1];
      d10 = __builtin_amdgcn_wmma_f32_16x16x4_f32(false, a, false, b, (short)0, d10, false, false);
      b.x = Sr1[c10]; b.y = Sr1[c10 + 1];
      d11 = __builtin_amdgcn_wmma_f32_16x16x4_f32(false, a, false, b, (short)0, d11, false, false);
    }
  }

  // Store: lane L, reg g of tile (yb,xb) -> out[yb + (L&15)][xb + g + (L>=16?8:0)]
  float* oc = out + (size_t)ch * 625;
  const int xo = (lane >> 4) * 8;

  {
    const int y = m;                      // yb = 0 (always < 25)
#pragma unroll
    for (int g = 0; g < 8; ++g) {
      int x = xo + g;                     // xb = 0
      if (x < 25) oc[y * 25 + x] = d00[g];
      x += 16;                            // xb = 16
      if (x < 25) oc[y * 25 + x] = d01[g];
    }
  }
  {
    const int y = 16 + m;                 // yb = 16
    if (y < 25) {
#pragma unroll
      for (int g = 0; g < 8; ++g) {
        int x = xo + g;
        if (x < 25) oc[y * 25 + x] = d10[g];
        x += 16;
        if (x < 25) oc[y * 25 + x] = d11[g];
      }
    }
  }
}

extern "C" void kernel_launch(void* const* d_in, const int* in_sizes, int n_in,
                              void* d_out, int out_size, void* d_ws, size_t ws_size,
                              hipStream_t stream) {
  const float* kern = (const float*)d_in[0];   // (B, C, 8, 8) fp32
  const float* srch = (const float*)d_in[1];   // (B, C, 32, 32) fp32
  float* out = (float*)d_out;                  // (B, C, 25, 25) fp32

  const int nchan = in_sizes[0] / 64;          // B*C = 32768
  const int blocks = (nchan + WAVES - 1) / WAVES;

  hipLaunchKernelGGL(xcorr_dw_wmma, dim3(blocks), dim3(WAVES * 32), 0, stream,
                     kern, srch, out, nchan);
}